// glocapBlock_12610023981573
// MI455X (gfx1250) — compile-verified
//
#include <hip/hip_runtime.h>

// glocaps dynamic routing, gfx1250 (wave32, WMMA).
// Key decision: never materialize v (256 MB); recompute it from l (16 MB,
// L2-resident) with V_WMMA_F32_16X16X4_F32 in every pass that needs it.

typedef __attribute__((ext_vector_type(2))) float v2f;
typedef __attribute__((ext_vector_type(8))) float v8f;

constexpr int kN    = 32;
constexpr int kCin  = 32;
constexpr int kCout = 16;
constexpr int kD    = 16;    // P*P
constexpr int kWH   = 256;   // W*H
constexpr int kM    = kCin * kWH;  // 8192
constexpr int ITERS = 3;           // reference num_iters
constexpr float kEps = 1e-6f;

// ---------------------------------------------------------------------------
// Compute a 16x64 tile of v for (n, c, sChunk) per wave and store it in LDS.
// Wave w handles s_local in [w*4, w*4+4), tile row M = (s_local%4)*4 + i.
// A (16x4 fp32, ISA layout): lanes 0-15 rows, K={0,1}; lanes 16-31 K={2,3}.
// B (4x16 fp32, mirrored layout): col N = o_local*4 + k per o-group.
// D (16x16 fp32, ISA layout): VGPR r -> row r (lanes<16) / r+8 (lanes>=16).
// ---------------------------------------------------------------------------
__device__ __forceinline__ void compute_v_tiles(
    const float* __restrict__ l, const float* __restrict__ wgt,
    int n, int c, int sChunk, float* __restrict__ ldsV /* [8][16][64] */)
{
    const int tid  = threadIdx.x;
    const int wave = tid >> 5;
    const int lane = tid & 31;
    const int row  = lane & 15;          // tile row / tile col index
    const bool hi  = lane >= 16;
    const int j0   = hi ? 2 : 0;         // K pair this half-wave owns

    // A: lp[n,c,s,i,j] = l[((n*Cin + c)*D + i*4+j)*WH + s]
    const int sLocal = wave * 4 + (row >> 2);
    const int iIdx   = row & 3;
    const int sGlob  = sChunk * 32 + sLocal;
    const float* lBase = l + ((size_t)(n * kCin + c) * kD) * kWH + sGlob;
    v2f A;
    A.x = lBase[(size_t)(iIdx * 4 + j0    ) * kWH];
    A.y = lBase[(size_t)(iIdx * 4 + j0 + 1) * kWH];

    // B: weight[c,o,j,k], flat = c*256 + o*16 + j*4 + k
    const int nCol   = row;              // column within an o-group (o_l*4+k)
    const int oLocal = nCol >> 2;
    const int kIdx   = nCol & 3;
    const float* wBase = wgt + (size_t)c * (kCout * 16);

    v8f acc[4];
#pragma unroll
    for (int og = 0; og < 4; ++og) {
        const int o = og * 4 + oLocal;
        v2f B;
        B.x = wBase[o * 16 + (j0    ) * 4 + kIdx];
        B.y = wBase[o * 16 + (j0 + 1) * 4 + kIdx];
        v8f z = {};
        // (neg_a, A, neg_b, B, c_mod, C, reuse_a, reuse_b)
        acc[og] = __builtin_amdgcn_wmma_f32_16x16x4_f32(
            false, A, false, B, (short)0, z, false, false);
    }

    // Scatter D tiles to LDS: tile[M][og*16 + N], col = o*4 + k overall.
    float* tile = ldsV + wave * (16 * 64);
#pragma unroll
    for (int r = 0; r < 8; ++r) {
        const int mRow = r + (hi ? 8 : 0);
#pragma unroll
        for (int og = 0; og < 4; ++og)
            tile[mRow * 64 + og * 16 + nCol] = acc[og][r];
    }
}

// ---------------------------------------------------------------------------
// init: gbuf = g0, sAcc = 0
// ---------------------------------------------------------------------------
__global__ __launch_bounds__(256)
void glocap_init_kernel(const float* __restrict__ g0,
                        float* __restrict__ gbuf, float* __restrict__ sAcc)
{
    const int t = blockIdx.x * 256 + threadIdx.x;
    if (t < kN * kCout * kD) {
        gbuf[t] = g0[t];
        sAcc[t] = 0.0f;
    }
}

// ---------------------------------------------------------------------------
// pass: b[n,m,o] (=|+=) sum_d v[n,m,o,d] * g[n,o,d]
// grid = N * Cin * 8 blocks (8 s-chunks of 32), 256 threads (8 waves).
// ---------------------------------------------------------------------------
__global__ __launch_bounds__(256)
void glocap_pass_kernel(const float* __restrict__ l, const float* __restrict__ wgt,
                        const float* __restrict__ gvec, float* __restrict__ b,
                        int accumulate)
{
    __shared__ float ldsV[8 * 16 * 64];   // 32 KB
    const int bx     = blockIdx.x;
    const int n      = bx / (kCin * 8);
    const int rem    = bx % (kCin * 8);
    const int c      = rem >> 3;
    const int sChunk = rem & 7;

    compute_v_tiles(l, wgt, n, c, sChunk, ldsV);
    __syncthreads();

    const float* g = gvec + (size_t)n * kCout * kD;
    for (int p = threadIdx.x; p < 32 * kCout; p += 256) {
        const int sLocal = p >> 4;
        const int o      = p & 15;
        const float* vb = ldsV + (sLocal >> 2) * 1024
                               + ((sLocal & 3) << 2) * 64 + o * 4;
        const float* gg = g + o * kD;
        float sum = 0.0f;
#pragma unroll
        for (int i = 0; i < 4; ++i)
#pragma unroll
            for (int k = 0; k < 4; ++k)
                sum += vb[i * 64 + k] * gg[i * 4 + k];

        const int sGlob = sChunk * 32 + sLocal;
        const size_t bIdx = ((size_t)n * kM + (size_t)c * kWH + sGlob) * kCout + o;
        if (accumulate) b[bIdx] += sum;
        else            b[bIdx] = sum;
    }
}

// ---------------------------------------------------------------------------
// stats: per (n,o) -> max_m b and sum_m exp(b-max) over M=8192
// grid = N*Cout blocks, 256 threads; 32 values per thread kept in registers.
// ---------------------------------------------------------------------------
__global__ __launch_bounds__(256)
void glocap_stats_kernel(const float* __restrict__ b,
                         float* __restrict__ mx, float* __restrict__ se)
{
    __shared__ float red[256];
    const int no = blockIdx.x;            // n*Cout + o
    const int n  = no >> 4;
    const int o  = no & 15;
    const int tid = threadIdx.x;

    const float* bBase = b + (size_t)n * kM * kCout + o;
    float vals[32];
    float m = -INFINITY;
#pragma unroll
    for (int q = 0; q < 32; ++q) {
        vals[q] = bBase[(size_t)(q * 256 + tid) * kCout];
        m = fmaxf(m, vals[q]);
    }
    red[tid] = m;
    __syncthreads();
    for (int st = 128; st > 0; st >>= 1) {
        if (tid < st) red[tid] = fmaxf(red[tid], red[tid + st]);
        __syncthreads();
    }
    const float gm = red[0];
    __syncthreads();

    float s = 0.0f;
#pragma unroll
    for (int q = 0; q < 32; ++q) s += __expf(vals[q] - gm);
    red[tid] = s;
    __syncthreads();
    for (int st = 128; st > 0; st >>= 1) {
        if (tid < st) red[tid] += red[tid + st];
        __syncthreads();
    }
    if (tid == 0) { mx[no] = gm; se[no] = red[0]; }
}

// ---------------------------------------------------------------------------
// route: cc = softmax(b); sAcc[n,o,d] += sum_s cc * v; optionally write a.
// Same tiling as pass; block-partial reduction in LDS, one atomicAdd per
// thread (thread = one (o,d) pair, 256 = 16*16 exactly).
// ---------------------------------------------------------------------------
__global__ __launch_bounds__(256)
void glocap_route_kernel(const float* __restrict__ l, const float* __restrict__ wgt,
                         const float* __restrict__ b, const float* __restrict__ mx,
                         const float* __restrict__ se, float* __restrict__ sAcc,
                         float* __restrict__ aOut, int writeA)
{
    __shared__ float ldsV[8 * 16 * 64];   // 32 KB
    __shared__ float ccBuf[32 * kCout];   // 2 KB
    const int bx     = blockIdx.x;
    const int n      = bx / (kCin * 8);
    const int rem    = bx % (kCin * 8);
    const int c      = rem >> 3;
    const int sChunk = rem & 7;

    compute_v_tiles(l, wgt, n, c, sChunk, ldsV);

    for (int p = threadIdx.x; p < 32 * kCout; p += 256) {
        const int sLocal = p >> 4;
        const int o      = p & 15;
        const int sGlob  = sChunk * 32 + sLocal;
        const size_t bIdx = ((size_t)n * kM + (size_t)c * kWH + sGlob) * kCout + o;
        const int no = n * kCout + o;
        const float cc = __expf(b[bIdx] - mx[no]) / se[no];
        ccBuf[sLocal * kCout + o] = cc;
        if (writeA) aOut[bIdx] = cc;      // a has same (n,m,o) layout as b
    }
    __syncthreads();

    const int o = threadIdx.x >> 4;
    const int d = threadIdx.x & 15;
    const int i = d >> 2;
    const int k = d & 3;
    float acc = 0.0f;
#pragma unroll 4
    for (int sLocal = 0; sLocal < 32; ++sLocal)
        acc += ccBuf[sLocal * kCout + o] *
               ldsV[(sLocal >> 2) * 1024 + (((sLocal & 3) << 2) + i) * 64 + o * 4 + k];
    atomicAdd(&sAcc[((size_t)n * kCout + o) * kD + d], acc);
}

// ---------------------------------------------------------------------------
// squash: g[n,o,:] = squash(sAcc[n,o,:]); zero sAcc; optionally write g out.
// ---------------------------------------------------------------------------
__global__ __launch_bounds__(256)
void glocap_squash_kernel(float* __restrict__ sAcc, float* __restrict__ gbuf,
                          float* __restrict__ gOut, int writeOut)
{
    const int t = blockIdx.x * 256 + threadIdx.x;   // (n*Cout + o)
    if (t >= kN * kCout) return;
    float* sp = sAcc + (size_t)t * kD;
    float v[kD];
    float n2 = 0.0f;
#pragma unroll
    for (int d = 0; d < kD; ++d) { v[d] = sp[d]; n2 += v[d] * v[d]; }
    const float coef = (n2 / (1.0f + n2)) / (sqrtf(n2) + kEps);
    float* gp = gbuf + (size_t)t * kD;
#pragma unroll
    for (int d = 0; d < kD; ++d) {
        const float gv = coef * v[d];
        gp[d] = gv;
        sp[d] = 0.0f;                                // ready for next iter
        if (writeOut) gOut[(size_t)t * kD + d] = gv;
    }
}

// ---------------------------------------------------------------------------
extern "C" void kernel_launch(void* const* d_in, const int* in_sizes, int n_in,
                              void* d_out, int out_size, void* d_ws, size_t ws_size,
                              hipStream_t stream)
{
    (void)in_sizes; (void)n_in; (void)out_size; (void)ws_size;
    const float* l   = (const float*)d_in[0];   // (N, Cin, D, W, H)
    const float* g0  = (const float*)d_in[1];   // (N, Cout, D)
    const float* wgt = (const float*)d_in[2];   // (1, Cin, 1, Cout, P, P)
    // d_in[3] = num_iters scalar; reference fixes it to 3 (ITERS).

    float* out  = (float*)d_out;
    float* aOut = out;                                 // (N, M, Cout)
    float* gOut = out + (size_t)kN * kM * kCout;       // (N, Cout, D)

    char* ws = (char*)d_ws;
    float* b    = (float*)ws; ws += (size_t)kN * kM * kCout * sizeof(float); // 16 MB
    float* gbuf = (float*)ws; ws += (size_t)kN * kCout * kD * sizeof(float);
    float* sAcc = (float*)ws; ws += (size_t)kN * kCout * kD * sizeof(float);
    float* mx   = (float*)ws; ws += (size_t)kN * kCout * sizeof(float);
    float* se   = (float*)ws; ws += (size_t)kN * kCout * sizeof(float);

    const int passGrid = kN * kCin * 8;   // 8192 blocks x 256 threads

    glocap_init_kernel<<<32, 256, 0, stream>>>(g0, gbuf, sAcc);
    glocap_pass_kernel<<<passGrid, 256, 0, stream>>>(l, wgt, gbuf, b, 0);

    for (int it = 0; it < ITERS; ++it) {
        const int last = (it == ITERS - 1);
        glocap_stats_kernel<<<kN * kCout, 256, 0, stream>>>(b, mx, se);
        glocap_route_kernel<<<passGrid, 256, 0, stream>>>(l, wgt, b, mx, se,
                                                          sAcc, aOut, last);
        glocap_squash_kernel<<<2, 256, 0, stream>>>(sAcc, gbuf, gOut, last);
        if (!last)  // final b-update is dead in the reference
            glocap_pass_kernel<<<passGrid, 256, 0, stream>>>(l, wgt, gbuf, b, 1);
    }
}